// CRFLayer_74363063763338
// MI455X (gfx1250) — compile-verified
//
#include <hip/hip_runtime.h>
#include <hip/hip_bf16.h>

typedef __attribute__((ext_vector_type(2))) float v2f;
typedef __attribute__((ext_vector_type(4))) float v4f;
typedef __attribute__((ext_vector_type(8))) float v8f;

#define TAGS 48
#define SEQ  1024
#define LPAD 20   // LDS row pad: 20n mod 64 distinct for n=0..15 (bank-clean)

__device__ __forceinline__ float half_max16(float u) {
  u = fmaxf(u, __shfl_xor(u, 1, 32));
  u = fmaxf(u, __shfl_xor(u, 2, 32));
  u = fmaxf(u, __shfl_xor(u, 4, 32));
  u = fmaxf(u, __shfl_xor(u, 8, 32));
  return u;
}
__device__ __forceinline__ float half_sum16(float u) {
  u += __shfl_xor(u, 1, 32);
  u += __shfl_xor(u, 2, 32);
  u += __shfl_xor(u, 4, 32);
  u += __shfl_xor(u, 8, 32);
  return u;
}

// One wave (32 lanes) handles 16 batch rows for the whole sequence.
// score kept in WMMA C-layout: sc[t][r] = score[batch = b0+r+8h][tag = 16t+n],
// h = lane>>4, n = lane&15.
__global__ void __launch_bounds__(32)
crf_forward_kernel(const float* __restrict__ emis,
                   const int*   __restrict__ mask,
                   const float* __restrict__ trans,
                   const float* __restrict__ startt,
                   const float* __restrict__ endt,
                   float*       __restrict__ part)
{
  __shared__ __align__(16) float plds[TAGS * LPAD]; // p column-major: plds[col*LPAD + row]
  const int lane = threadIdx.x & 31;
  const int h = lane >> 4;
  const int n = lane & 15;
  const int b0 = blockIdx.x * 16;

  // E = exp(transitions), resident in registers in WMMA B-layout:
  // chunk c (K rows 4c..4c+3), tile t: v0 = E[4c+2h][16t+n], v1 = E[4c+2h+1][16t+n]
  v2f Bop[12][3];
#pragma unroll
  for (int c = 0; c < 12; ++c) {
    const int row = 4 * c + 2 * h;
#pragma unroll
    for (int t = 0; t < 3; ++t) {
      const int col = 16 * t + n;
      v2f bb;
      bb.x = __expf(trans[row * TAGS + col]);
      bb.y = __expf(trans[(row + 1) * TAGS + col]);
      Bop[c][t] = bb;
    }
  }

  // score0 = start_transitions + emissions[:, 0]
  v8f sc[3];
#pragma unroll
  for (int t = 0; t < 3; ++t) {
    const int col = 16 * t + n;
    const float st = startt[col];
#pragma unroll
    for (int r = 0; r < 8; ++r) {
      const int b = b0 + r + 8 * h;
      sc[t][r] = st + emis[(size_t)b * SEQ * TAGS + col];
    }
  }

  // prefetch emissions for s = 1
  v8f em[3];
#pragma unroll
  for (int t = 0; t < 3; ++t)
#pragma unroll
    for (int r = 0; r < 8; ++r) {
      const int b = b0 + r + 8 * h;
      em[t][r] = emis[((size_t)b * SEQ + 1) * TAGS + 16 * t + n];
    }

  for (int s = 1; s < SEQ; ++s) {
    // Prefetch next step's emissions (clamped index: uniform trip count, no
    // loop-body duplication; last iteration's loads are dead) and this step's
    // mask. Issued first so they get the whole step's latency.
    const int sp = (s + 1 < SEQ) ? s + 1 : SEQ - 1;
    v8f em2[3];
#pragma unroll
    for (int t = 0; t < 3; ++t)
#pragma unroll
      for (int r = 0; r < 8; ++r) {
        const int b = b0 + r + 8 * h;
        em2[t][r] = emis[((size_t)b * SEQ + sp) * TAGS + 16 * t + n];
      }
    int mk[8];
#pragma unroll
    for (int r = 0; r < 8; ++r)
      mk[r] = mask[(size_t)(b0 + r + 8 * h) * SEQ + s];

    // Normalizer: column-0 score of each row (exact shift for logsumexp; keeps
    // exp args bounded ~ +/-25, no overflow/underflow). One broadcast per row.
    float mx[8];
#pragma unroll
    for (int r = 0; r < 8; ++r)
      mx[r] = __shfl(sc[0][r], 0, 16);

    // p = exp(score - mx) -> LDS, column-major; 8 contiguous rows per lane,
    // stored as two b128 stores per tile.
#pragma unroll
    for (int t = 0; t < 3; ++t) {
      v4f w0, w1;
      w0.x = __expf(sc[t][0] - mx[0]);
      w0.y = __expf(sc[t][1] - mx[1]);
      w0.z = __expf(sc[t][2] - mx[2]);
      w0.w = __expf(sc[t][3] - mx[3]);
      w1.x = __expf(sc[t][4] - mx[4]);
      w1.y = __expf(sc[t][5] - mx[5]);
      w1.z = __expf(sc[t][6] - mx[6]);
      w1.w = __expf(sc[t][7] - mx[7]);
      float* dst = &plds[(16 * t + n) * LPAD + 8 * h];
      *(v4f*)dst = w0;
      *(v4f*)(dst + 4) = w1;
    }
    __syncthreads();

    // reload as A-layout chunks: lane l -> M=n, K = 4c+2h, 4c+2h+1
    v2f A[12];
#pragma unroll
    for (int c = 0; c < 12; ++c) {
      v2f a;
      a.x = plds[(4 * c + 2 * h) * LPAD + n];
      a.y = plds[(4 * c + 2 * h + 1) * LPAD + n];
      A[c] = a;
    }
    __syncthreads();

    // V = P @ E : two independent accumulation chains per tile (dep distance 6)
    const v8f z = {0.f, 0.f, 0.f, 0.f, 0.f, 0.f, 0.f, 0.f};
    v8f accA[3] = {z, z, z};
    v8f accB[3] = {z, z, z};
#pragma unroll
    for (int cc = 0; cc < 6; ++cc) {
      const int c0 = 2 * cc, c1 = 2 * cc + 1;
#pragma unroll
      for (int t = 0; t < 3; ++t)
        accA[t] = __builtin_amdgcn_wmma_f32_16x16x4_f32(
            false, A[c0], false, Bop[c0][t], (short)0, accA[t], false, false);
#pragma unroll
      for (int t = 0; t < 3; ++t)
        accB[t] = __builtin_amdgcn_wmma_f32_16x16x4_f32(
            false, A[c1], false, Bop[c1][t], (short)0, accB[t], false, false);
    }

    // score' = mx + log(V) + emissions; masked update
#pragma unroll
    for (int t = 0; t < 3; ++t) {
      const v8f vsum = accA[t] + accB[t];
#pragma unroll
      for (int r = 0; r < 8; ++r) {
        const float nv = mx[r] + __logf(vsum[r]) + em[t][r];
        sc[t][r] = (mk[r] > 0) ? nv : sc[t][r];
      }
    }

#pragma unroll
    for (int t = 0; t < 3; ++t) em[t] = em2[t];
  }

  // partition = logsumexp(score + end_transitions) per batch row (exact max here)
#pragma unroll
  for (int t = 0; t < 3; ++t) {
    const float e = endt[16 * t + n];
#pragma unroll
    for (int r = 0; r < 8; ++r) sc[t][r] += e;
  }
  float res[8];
#pragma unroll
  for (int r = 0; r < 8; ++r) {
    float u = fmaxf(sc[0][r], fmaxf(sc[1][r], sc[2][r]));
    const float m = half_max16(u);
    float sm = __expf(sc[0][r] - m) + __expf(sc[1][r] - m) + __expf(sc[2][r] - m);
    sm = half_sum16(sm);
    res[r] = m + __logf(sm);
  }
  if (n == 0) {
#pragma unroll
    for (int r = 0; r < 8; ++r)
      part[b0 + r + 8 * h] = res[r];
  }
}

// Gold path score: one wave per batch, lane-strided over the sequence.
__global__ void __launch_bounds__(256)
crf_gold_kernel(const float* __restrict__ emis, const int* __restrict__ tags,
                const int* __restrict__ mask, const float* __restrict__ trans,
                const float* __restrict__ startt, const float* __restrict__ endt,
                float* __restrict__ gold)
{
  const int wid = threadIdx.x >> 5;
  const int lane = threadIdx.x & 31;
  const int b = blockIdx.x * 8 + wid;
  const int* tg = tags + (size_t)b * SEQ;
  const int* mkp = mask + (size_t)b * SEQ;
  float acc = 0.f;
  int cnt = 0;
  for (int s = lane; s < SEQ; s += 32) {
    const int tc = tg[s];
    const int m = mkp[s];
    cnt += m;
    if (s == 0) {
      acc += startt[tc] + emis[(size_t)b * SEQ * TAGS + tc];
    } else {
      const int tp = tg[s - 1];
      acc += (trans[tp * TAGS + tc] +
              emis[((size_t)b * SEQ + s) * TAGS + tc]) * (float)m;
    }
  }
  for (int off = 16; off; off >>= 1) {
    acc += __shfl_xor(acc, off, 32);
    cnt += __shfl_xor(cnt, off, 32);
  }
  if (lane == 0) gold[b] = acc + endt[tg[cnt - 1]];
}

// loss = mean(part - gold)
__global__ void __launch_bounds__(256)
crf_reduce_kernel(const float* __restrict__ part, const float* __restrict__ gold,
                  float* __restrict__ out, int B)
{
  __shared__ float red[256];
  float v = 0.f;
  for (int i = threadIdx.x; i < B; i += 256) v += part[i] - gold[i];
  red[threadIdx.x] = v;
  __syncthreads();
  for (int off = 128; off > 0; off >>= 1) {
    if ((int)threadIdx.x < off) red[threadIdx.x] += red[threadIdx.x + off];
    __syncthreads();
  }
  if (threadIdx.x == 0) out[0] = red[0] / (float)B;
}

extern "C" void kernel_launch(void* const* d_in, const int* in_sizes, int n_in,
                              void* d_out, int out_size, void* d_ws, size_t ws_size,
                              hipStream_t stream) {
  const float* emis   = (const float*)d_in[0];
  const int*   tags   = (const int*)d_in[1];
  const int*   mask   = (const int*)d_in[2];
  const float* trans  = (const float*)d_in[3];
  const float* startt = (const float*)d_in[4];
  const float* endt   = (const float*)d_in[5];
  const int B = in_sizes[1] / SEQ;  // 512

  float* wsf  = (float*)d_ws;
  float* part = wsf;
  float* gold = wsf + B;

  hipLaunchKernelGGL(crf_forward_kernel, dim3(B / 16), dim3(32), 0, stream,
                     emis, mask, trans, startt, endt, part);
  hipLaunchKernelGGL(crf_gold_kernel, dim3(B / 8), dim3(256), 0, stream,
                     emis, tags, mask, trans, startt, endt, gold);
  hipLaunchKernelGGL(crf_reduce_kernel, dim3(1), dim3(256), 0, stream,
                     part, gold, (float*)d_out, B);
}